// SpaceDepthSepConv2_42314017800975
// MI455X (gfx1250) — compile-verified
//
#include <hip/hip_runtime.h>

typedef __attribute__((ext_vector_type(2))) float v2f;
typedef __attribute__((ext_vector_type(8))) float v8f;

#define N_ 64
#define H_ 256
#define W_ 256
#define C_ 16
#define O_ 16
#define TILE_H 16
#define PSTR 18              // padded per-pixel LDS stride (floats) -> conflict-free lane halves
#define ROWSZ (W_ * PSTR)    // floats per buffered row

__global__ __launch_bounds__(512)
void SpaceDepthSepConv2_cp_fused(const float* __restrict__ in,
                                 const float* __restrict__ u,
                                 const float* __restrict__ v,
                                 const float* __restrict__ w,
                                 const float* __restrict__ b,
                                 float* __restrict__ out)
{
    __shared__ float ring[3 * ROWSZ];   // 54 KB: 3-row ring of channel-mixed rows t[x][l]

    const int tid  = threadIdx.x;       // 0..511 (16 wave32s)
    const int wave = tid >> 5;
    const int lane = tid & 31;
    const int p    = lane & 15;         // pixel-in-tile (WMMA M row)
    const int half = lane >> 4;         // lane half selects K pair per ISA A/B layout
    const int x0   = wave << 4;         // 16-pixel column tile per wave

    const int bid   = blockIdx.x;
    const int n     = bid / (H_ / TILE_H);
    const int strip = bid % (H_ / TILE_H);
    const int y0    = strip * TILE_H;

    // B matrix = w^T (input-ch x out-ch), 4 K-chunks for 16x16x4 f32 WMMA.
    // B[k][n]: VGPR0 holds K=4kc+2*half, VGPR1 holds K=4kc+2*half+1, N = lane&15.
    v2f bmat[4];
    {
        const int l  = p;          // output channel index (N)
        const int kb = 2 * half;   // K sub-offset for this lane half
        #pragma unroll
        for (int kc = 0; kc < 4; ++kc) {
            bmat[kc].x = w[l * C_ + 4 * kc + kb];
            bmat[kc].y = w[l * C_ + 4 * kc + kb + 1];
        }
    }

    // Output-stage mapping: thread -> fixed out-channel oc, pixels x = xbase + 32k.
    const int oc    = tid & 15;
    const int xbase = tid >> 4;    // 0..31
    const float u0 = u[oc * 3 + 0], u1 = u[oc * 3 + 1], u2 = u[oc * 3 + 2];
    const float v0 = v[oc * 3 + 0], v1 = v[oc * 3 + 1], v2 = v[oc * 3 + 2];
    const float bias = b[oc];

    for (int step = 0; step < TILE_H + 2; ++step) {
        const int ty = y0 - 1 + step;                 // t row being produced
        float* trow = &ring[(step % 3) * ROWSZ];

        // ---- Stage 1: channel mix t[x][l] = sum_c in[x][c] * w[l][c] via f32 WMMA ----
        v8f acc = {};
        if (ty >= 0 && ty < H_) {                     // uniform per block: EXEC stays all-1s
            const float* src = in
                + ((((size_t)n * H_ + ty) * W_) + x0 + p) * C_ + 2 * half;
            #pragma unroll
            for (int kc = 0; kc < 4; ++kc) {
                v2f a = *(const v2f*)(src + 4 * kc);  // channels {4kc+2h, 4kc+2h+1} of pixel p
                acc = __builtin_amdgcn_wmma_f32_16x16x4_f32(
                    false, a, false, bmat[kc], (short)0, acc, false, false);
            }
        }
        // C/D layout: lane holds N = lane&15, VGPR r -> M = r + 8*half
        {
            const int px = x0 + 8 * half;
            const int cl = lane & 15;
            #pragma unroll
            for (int r = 0; r < 8; ++r)
                trow[(px + r) * PSTR + cl] = acc[r];  // zeros for out-of-range rows (SAME pad)
        }

        __syncthreads();

        // ---- Stage 2: separable 3x3 taps + bias + ReLU once 3 t-rows are live ----
        if (step >= 2) {
            const int oy = ty - 1;
            const float* t0 = &ring[((step - 2) % 3) * ROWSZ];  // row oy-1  (v[l,0])
            const float* t1 = &ring[((step - 1) % 3) * ROWSZ];  // row oy    (v[l,1])
            const float* t2 = trow;                             // row oy+1  (v[l,2])
            #pragma unroll
            for (int k = 0; k < 8; ++k) {
                const int x = xbase + 32 * k;
                float h0 = u1 * t0[x * PSTR + oc];
                float h1 = u1 * t1[x * PSTR + oc];
                float h2 = u1 * t2[x * PSTR + oc];
                if (x > 0) {
                    h0 = fmaf(u0, t0[(x - 1) * PSTR + oc], h0);
                    h1 = fmaf(u0, t1[(x - 1) * PSTR + oc], h1);
                    h2 = fmaf(u0, t2[(x - 1) * PSTR + oc], h2);
                }
                if (x < W_ - 1) {
                    h0 = fmaf(u2, t0[(x + 1) * PSTR + oc], h0);
                    h1 = fmaf(u2, t1[(x + 1) * PSTR + oc], h1);
                    h2 = fmaf(u2, t2[(x + 1) * PSTR + oc], h2);
                }
                float o = fmaf(v0, h0, fmaf(v1, h1, fmaf(v2, h2, bias)));
                out[(((size_t)n * H_ + oy) * W_ + x) * C_ + oc] = fmaxf(o, 0.0f);
            }
        }
        __syncthreads();   // ring slot reused next iteration
    }
}

extern "C" void kernel_launch(void* const* d_in, const int* in_sizes, int n_in,
                              void* d_out, int out_size, void* d_ws, size_t ws_size,
                              hipStream_t stream) {
    (void)in_sizes; (void)n_in; (void)d_ws; (void)ws_size; (void)out_size;
    const float* in = (const float*)d_in[0];
    const float* u  = (const float*)d_in[1];
    const float* v  = (const float*)d_in[2];
    const float* w  = (const float*)d_in[3];
    const float* b  = (const float*)d_in[4];
    float* out = (float*)d_out;

    dim3 grid(N_ * (H_ / TILE_H));   // 64 images x 16 row-strips = 1024 blocks
    dim3 block(512);                 // 16 wave32s; each wave owns a 16-pixel WMMA tile
    SpaceDepthSepConv2_cp_fused<<<grid, block, 0, stream>>>(in, u, v, w, b, out);
}